// Attention_84602265797043
// MI455X (gfx1250) — compile-verified
//
#include <hip/hip_runtime.h>
#include <hip/hip_bf16.h>
#include <math.h>

typedef __attribute__((ext_vector_type(2))) float v2f;
typedef __attribute__((ext_vector_type(8))) float v8f;

// D = A(16x4) * B(4x16) + C, all f32.  Layout assumption (ISA 7.12.2):
//   A: lane l, vgpr v -> A[l%16][v + 2*(l/16)]
//   B: lane l, vgpr v -> B[v + 2*(l/16)][l%16]
//   D: vgpr r, lane l -> D[r + 8*(l/16)][l%16]
__device__ __forceinline__ v8f wmma4(v2f a, v2f b, v8f c) {
  return __builtin_amdgcn_wmma_f32_16x16x4_f32(false, a, false, b, (short)0, c,
                                               false, false);
}

// ---------------- Kernel 1: corr (adj) + e0 (-> ef[b][c][m]) ----------------
__global__ void k1_corr_e0(const float* __restrict__ x,
                           float* __restrict__ adj,
                           float* __restrict__ ef) {
  __shared__ float xs[32 * 256];   // raw x tile
  __shared__ float xc[32 * 256];   // centered
  __shared__ float rowsum[32];
  __shared__ float cov[32 * 32];
  __shared__ float nrm[32];

  const int bm = blockIdx.x;              // 0..95
  const int b = bm / 12, m = bm % 12;
  const int t = threadIdx.x;              // 0..255

  const float* xsrc = x + (size_t)bm * 8192;
  for (int i = 0; i < 32; ++i) xs[t + 256 * i] = xsrc[t + 256 * i];
  __syncthreads();

  if (t < 32) {
    float s = 0.f;
    for (int d = 0; d < 256; ++d) s += xs[t * 256 + d];
    rowsum[t] = s;
  }
  __syncthreads();

  for (int i = 0; i < 32; ++i) {
    int idx = t + 256 * i;
    xc[idx] = xs[idx] - rowsum[idx >> 8] * (1.f / 256.f);
  }
  __syncthreads();

  const int w = t >> 5, lane = t & 31;
  const int lm = lane & 15, lh = lane >> 4;

  // cov = xc (32x256) * xc^T : 4 tiles of 16x16, one per wave 0..3
  if (w < 4) {
    const int mt = w >> 1, nt = w & 1;
    v8f acc = {};
    for (int k0 = 0; k0 < 64; ++k0) {
      const int kb = 4 * k0 + 2 * lh;
      v2f a, bb;
      a.x = xc[(mt * 16 + lm) * 256 + kb];
      a.y = xc[(mt * 16 + lm) * 256 + kb + 1];
      bb.x = xc[(nt * 16 + lm) * 256 + kb];   // B[k][n] = xc[n][k]
      bb.y = xc[(nt * 16 + lm) * 256 + kb + 1];
      acc = wmma4(a, bb, acc);
    }
    for (int r = 0; r < 8; ++r) {
      const int row = mt * 16 + r + 8 * lh;
      cov[row * 32 + nt * 16 + lm] = acc[r];
    }
  }
  __syncthreads();

  if (t < 32) nrm[t] = sqrtf(fmaxf(cov[t * 33], 1e-12f));
  __syncthreads();

  for (int j = 0; j < 4; ++j) {
    const int p = t + 256 * j;           // c = n*32 + k
    const int n = p >> 5, k = p & 31;
    adj[(size_t)bm * 1024 + p] = cov[p] / (nrm[n] * nrm[k]);

    // sum_d pos*x[k,d] = 2*rowsum[k] - sum_d x[k,d]*[x[n,d]==x[k,d]]
    float eq = 0.f;
    for (int d = 0; d < 256; ++d) {
      const float xv = xs[k * 256 + d];
      eq += (xs[n * 256 + d] == xv) ? xv : 0.f;
    }
    const float e0 = (rowsum[n] + 2.f * rowsum[k] - eq) * (1.f / 512.f);
    ef[((size_t)b * 1024 + p) * 12 + m] = e0;
  }
}

// ---------------- Kernel 2: h = relu(W1 (512x1024) @ ef_b (1024x12)) --------
__global__ void k2_mlp1(const float* __restrict__ W1,
                        const float* __restrict__ ef,
                        float* __restrict__ h) {
  __shared__ float efs[1024 * 12];   // 48 KB
  const int b = blockIdx.x;
  const int t = threadIdx.x;
  const float* src = ef + (size_t)b * 12288;
  for (int i = 0; i < 48; ++i) efs[t + 256 * i] = src[t + 256 * i];
  __syncthreads();

  const int w = t >> 5, lane = t & 31, lm = lane & 15, lh = lane >> 4;
  for (int j = 0; j < 4; ++j) {
    const int rt = w + 8 * j;          // 32 row tiles of 16
    const int row = rt * 16 + lm;
    v8f acc = {};
    for (int k0 = 0; k0 < 256; ++k0) {
      const int kb = 4 * k0 + 2 * lh;
      v2f a, bb;
      a.x = W1[(size_t)row * 1024 + kb];       // consecutive -> b64 load
      a.y = W1[(size_t)row * 1024 + kb + 1];
      bb.x = (lm < 12) ? efs[kb * 12 + lm] : 0.f;       // pad N 12->16
      bb.y = (lm < 12) ? efs[(kb + 1) * 12 + lm] : 0.f;
      acc = wmma4(a, bb, acc);
    }
    if (lm < 12) {
      for (int r = 0; r < 8; ++r) {
        const int orow = rt * 16 + r + 8 * lh;
        h[((size_t)b * 512 + orow) * 12 + lm] = fmaxf(acc[r], 0.f);
      }
    }
  }
}

// ------- Kernel 3: e = sigmoid(W2 (1024x512) @ h_b (512x12)) -> [bm][c] -----
__global__ void k3_mlp2(const float* __restrict__ W2,
                        const float* __restrict__ h,
                        float* __restrict__ e) {
  __shared__ float hs[512 * 12];     // 24 KB
  const int b = blockIdx.x;
  const int t = threadIdx.x;
  const float* src = h + (size_t)b * 6144;
  for (int i = 0; i < 24; ++i) hs[t + 256 * i] = src[t + 256 * i];
  __syncthreads();

  const int w = t >> 5, lane = t & 31, lm = lane & 15, lh = lane >> 4;
  for (int j = 0; j < 8; ++j) {
    const int rt = w + 8 * j;          // 64 row tiles of 16
    const int row = rt * 16 + lm;
    v8f acc = {};
    for (int k0 = 0; k0 < 128; ++k0) {
      const int kb = 4 * k0 + 2 * lh;
      v2f a, bb;
      a.x = W2[(size_t)row * 512 + kb];
      a.y = W2[(size_t)row * 512 + kb + 1];
      bb.x = (lm < 12) ? hs[kb * 12 + lm] : 0.f;
      bb.y = (lm < 12) ? hs[(kb + 1) * 12 + lm] : 0.f;
      acc = wmma4(a, bb, acc);
    }
    if (lm < 12) {
      for (int r = 0; r < 8; ++r) {
        const int c = rt * 16 + r + 8 * lh;              // 0..1023
        const float v = acc[r];
        e[((size_t)(b * 12 + lm)) * 1024 + c] = 1.f / (1.f + expf(-v));
      }
    }
  }
}

// ------- Kernel 4: mask + softmax + out = att (32x32) @ x (32x256) ----------
__global__ void k4_attn(const float* __restrict__ x,
                        const float* __restrict__ adj,
                        const float* __restrict__ e,
                        float* __restrict__ out) {
  __shared__ float xs[32 * 256];
  __shared__ float att[32 * 32];
  const int bm = blockIdx.x;
  const int t = threadIdx.x;

  const float* xsrc = x + (size_t)bm * 8192;
  for (int i = 0; i < 32; ++i) xs[t + 256 * i] = xsrc[t + 256 * i];
  for (int j = 0; j < 4; ++j) {
    const int p = t + 256 * j;
    const float a = adj[(size_t)bm * 1024 + p];
    const float ev = e[(size_t)bm * 1024 + p];
    att[p] = (a > 0.f) ? ev : -1e12f;
  }
  __syncthreads();

  if (t < 32) {                        // row-wise softmax
    float mx = att[t * 32];
    for (int k = 1; k < 32; ++k) mx = fmaxf(mx, att[t * 32 + k]);
    float s = 0.f;
    for (int k = 0; k < 32; ++k) {
      const float ex = expf(att[t * 32 + k] - mx);
      att[t * 32 + k] = ex;
      s += ex;
    }
    const float inv = 1.f / s;
    for (int k = 0; k < 32; ++k) att[t * 32 + k] *= inv;
  }
  __syncthreads();

  const int w = t >> 5, lane = t & 31, lm = lane & 15, lh = lane >> 4;
  for (int j = 0; j < 4; ++j) {
    const int tile = w + 8 * j;        // 0..31 : mt in {0,1}, nt in 0..15
    const int mt = tile >> 4, nt = tile & 15;
    v8f acc = {};
    for (int k0 = 0; k0 < 8; ++k0) {
      const int kb = 4 * k0 + 2 * lh;
      v2f a, bb;
      a.x = att[(mt * 16 + lm) * 32 + kb];
      a.y = att[(mt * 16 + lm) * 32 + kb + 1];
      bb.x = xs[kb * 256 + nt * 16 + lm];
      bb.y = xs[(kb + 1) * 256 + nt * 16 + lm];
      acc = wmma4(a, bb, acc);
    }
    for (int r = 0; r < 8; ++r) {
      const int orow = mt * 16 + r + 8 * lh;
      out[(size_t)bm * 8192 + orow * 256 + nt * 16 + lm] = acc[r];
    }
  }
}

extern "C" void kernel_launch(void* const* d_in, const int* in_sizes, int n_in,
                              void* d_out, int out_size, void* d_ws,
                              size_t ws_size, hipStream_t stream) {
  (void)in_sizes; (void)n_in; (void)out_size; (void)ws_size;
  const float* x  = (const float*)d_in[0];   // (8,12,32,256)
  const float* W1 = (const float*)d_in[1];   // (512,1024)
  const float* W2 = (const float*)d_in[2];   // (1024,512)
  float* out = (float*)d_out;                // (8,12,32,256)

  float* ws   = (float*)d_ws;
  float* adj  = ws;                 // 96*1024
  float* ef   = ws + 98304;         // 8*1024*12
  float* h    = ws + 196608;        // 8*512*12
  float* e    = ws + 245760;        // 96*1024  (total ~1.35 MB)

  k1_corr_e0<<<96, 256, 0, stream>>>(x, adj, ef);
  k2_mlp1  <<<8,  256, 0, stream>>>(W1, ef, h);
  k3_mlp2  <<<8,  256, 0, stream>>>(W2, h, e);
  k4_attn  <<<96, 256, 0, stream>>>(x, adj, e, out);
}